// GCN_11991548690782
// MI455X (gfx1250) — compile-verified
//
#include <hip/hip_runtime.h>

typedef __attribute__((ext_vector_type(2))) float v2f;
typedef __attribute__((ext_vector_type(8))) float v8f;

#define NN 100000
#define NE 3200000
#define DF 128
#define HD 32
#define NC 2
#define NTILES (NN / 16)   // 6250, exact — no row tail

// ---------------- degree / normalization ----------------

__global__ __launch_bounds__(256) void k_deg_init(float* __restrict__ deg) {
    int i = blockIdx.x * 256 + threadIdx.x;
    if (i < NN) deg[i] = 1.0f;            // self-loop contributes 1 to every node
}

__global__ __launch_bounds__(256) void k_deg_edge(const int* __restrict__ dst,
                                                  float* __restrict__ deg) {
    int t = blockIdx.x * 256 + threadIdx.x;
    int stride = gridDim.x * 256;
    for (int e = t; e < NE; e += stride)
        atomicAdd(&deg[dst[e]], 1.0f);
}

__global__ __launch_bounds__(256) void k_dinv(float* __restrict__ deg) {
    int i = blockIdx.x * 256 + threadIdx.x;
    if (i < NN) deg[i] = rsqrtf(deg[i]);  // deg >= 1 always (self loops)
}

// ---------------- layer-1 GEMM: H = X @ W1 via f32 WMMA ----------------
// Per wave: one 16-row node tile, full 32 hidden cols (two 16x16 D tiles).
// A 16x4 f32 frag: lane L holds X[tile*16 + (L&15)][k0 + 2*(L>=16) + {0,1}]
// B 4x16  f32 frag: lane L holds W1[k0 + 2*(L>=16) + {0,1}][n0 + (L&15)]
// D 16x16 f32:      lane L, vgpr v holds D[v + 8*(L>=16)][L&15]

__global__ __launch_bounds__(256) void k_gemm1(const float* __restrict__ X,
                                               const float* __restrict__ W1,
                                               float* __restrict__ H) {
    __shared__ float sw[DF * HD];         // 16 KB staged W1
    {
        const float4* w4 = (const float4*)W1;
        float4* s4 = (float4*)sw;
        for (int i = threadIdx.x; i < (DF * HD) / 4; i += 256) s4[i] = w4[i];
    }
    __syncthreads();

    const int wave = threadIdx.x >> 5;           // 0..7 (wave32)
    const int lane = threadIdx.x & 31;
    const int tile = blockIdx.x * 8 + wave;      // 16-row node tile
    if (tile >= NTILES) return;                  // wave-uniform: EXEC stays all-1 for WMMA

    const int hi = lane >> 4;                    // 0 or 1
    const int nl = lane & 15;

    const float* __restrict__ xrow = X + (size_t)(tile * 16 + nl) * DF;

    v8f acc0 = {};
    v8f acc1 = {};

    #pragma unroll 4
    for (int k0 = 0; k0 < DF; k0 += 4) {
        const int ka = k0 + (hi << 1);
        v2f a;
        a.x = xrow[ka];
        a.y = xrow[ka + 1];

        v2f b0, b1v;
        b0.x  = sw[ka * HD + nl];
        b0.y  = sw[(ka + 1) * HD + nl];
        b1v.x = sw[ka * HD + 16 + nl];
        b1v.y = sw[(ka + 1) * HD + 16 + nl];

        acc0 = __builtin_amdgcn_wmma_f32_16x16x4_f32(false, a, false, b0,
                                                     (short)0, acc0, false, false);
        acc1 = __builtin_amdgcn_wmma_f32_16x16x4_f32(false, a, false, b1v,
                                                     (short)0, acc1, false, false);
    }

    const size_t mbase = (size_t)tile * 16;
    #pragma unroll
    for (int v = 0; v < 8; ++v) {
        const int row = v + (hi << 3);
        H[(mbase + row) * HD + nl]      = acc0[v];
        H[(mbase + row) * HD + 16 + nl] = acc1[v];
    }
}

// ---------------- generic zero ----------------

__global__ __launch_bounds__(256) void k_zero(float* __restrict__ p, int n) {
    int i = blockIdx.x * 256 + threadIdx.x;
    if (i < n) p[i] = 0.0f;
}

// ---------------- layer-1 scatter: wave per edge, lane per feature ----------------

__global__ __launch_bounds__(256) void k_scatter1(const int* __restrict__ src,
                                                  const int* __restrict__ dst,
                                                  const float* __restrict__ dinv,
                                                  const float* __restrict__ H,
                                                  float* __restrict__ AGG) {
    const int lane = threadIdx.x & 31;
    int wid = (blockIdx.x * 256 + threadIdx.x) >> 5;
    const int nwaves = (gridDim.x * 256) >> 5;
    for (int e = wid; e < NE; e += nwaves) {
        const int s = src[e];
        const int d = dst[e];
        const float nrm = dinv[s] * dinv[d];
        const float val = H[(size_t)s * HD + lane] * nrm;
        atomicAdd(&AGG[(size_t)d * HD + lane], val);
    }
}

// self-loop term + bias + ReLU, in place: AGG1 -> H1
__global__ __launch_bounds__(256) void k_relu_bias(float* __restrict__ AGG1,
                                                   const float* __restrict__ H,
                                                   const float* __restrict__ dinv,
                                                   const float* __restrict__ b1) {
    int t = blockIdx.x * 256 + threadIdx.x;
    if (t >= NN * HD) return;
    const int i = t >> 5;
    const int f = t & 31;
    const float di = dinv[i];
    const float v = AGG1[t] + H[t] * (di * di) + b1[f];
    AGG1[t] = v > 0.0f ? v : 0.0f;
}

// ---------------- layer-2 GEMM (tiny): G = H1 @ W2 ----------------

__global__ __launch_bounds__(256) void k_gemm2(const float* __restrict__ H1,
                                               const float* __restrict__ W2,
                                               float* __restrict__ G) {
    int i = blockIdx.x * 256 + threadIdx.x;
    if (i >= NN) return;
    const float* __restrict__ r = H1 + (size_t)i * HD;
    float g0 = 0.0f, g1 = 0.0f;
    #pragma unroll
    for (int k = 0; k < HD; ++k) {
        const float h = r[k];
        g0 = fmaf(h, W2[k * 2], g0);
        g1 = fmaf(h, W2[k * 2 + 1], g1);
    }
    G[(size_t)i * 2]     = g0;
    G[(size_t)i * 2 + 1] = g1;
}

// ---------------- layer-2 scatter: thread per edge, 2 atomics ----------------

__global__ __launch_bounds__(256) void k_scatter2(const int* __restrict__ src,
                                                  const int* __restrict__ dst,
                                                  const float* __restrict__ dinv,
                                                  const float* __restrict__ G,
                                                  float* __restrict__ AGG2) {
    int e = blockIdx.x * 256 + threadIdx.x;
    if (e >= NE) return;
    const int s = src[e];
    const int d = dst[e];
    const float nrm = dinv[s] * dinv[d];
    atomicAdd(&AGG2[(size_t)d * 2],     G[(size_t)s * 2]     * nrm);
    atomicAdd(&AGG2[(size_t)d * 2 + 1], G[(size_t)s * 2 + 1] * nrm);
}

// self-loop + bias + log_softmax over 2 classes
__global__ __launch_bounds__(256) void k_out(const float* __restrict__ AGG2,
                                             const float* __restrict__ G,
                                             const float* __restrict__ dinv,
                                             const float* __restrict__ b2,
                                             float* __restrict__ out) {
    int i = blockIdx.x * 256 + threadIdx.x;
    if (i >= NN) return;
    const float di = dinv[i];
    const float sl = di * di;
    const float l0 = AGG2[(size_t)i * 2]     + G[(size_t)i * 2]     * sl + b2[0];
    const float l1 = AGG2[(size_t)i * 2 + 1] + G[(size_t)i * 2 + 1] * sl + b2[1];
    const float m = fmaxf(l0, l1);
    const float lse = m + logf(expf(l0 - m) + expf(l1 - m));
    out[(size_t)i * 2]     = l0 - lse;
    out[(size_t)i * 2 + 1] = l1 - lse;
}

// ---------------- launcher ----------------

extern "C" void kernel_launch(void* const* d_in, const int* in_sizes, int n_in,
                              void* d_out, int out_size, void* d_ws, size_t ws_size,
                              hipStream_t stream) {
    const float* X  = (const float*)d_in[0];
    const int*   EI = (const int*)d_in[1];   // int32 per harness convention
    const float* W1 = (const float*)d_in[2];
    const float* b1 = (const float*)d_in[3];
    const float* W2 = (const float*)d_in[4];
    const float* b2 = (const float*)d_in[5];
    float* out = (float*)d_out;

    float* ws   = (float*)d_ws;
    float* dinv = ws;                          // NN floats (deg, then rsqrt in place)
    float* H    = ws + NN;                     // NN*32
    float* AGG1 = ws + (size_t)NN * 33;        // NN*32 (becomes H1 after relu)
    float* G    = ws + (size_t)NN * 65;        // NN*2
    float* AGG2 = ws + (size_t)NN * 67;        // NN*2   (total 69*NN*4 B ≈ 27.6 MB)

    const int* src = EI;
    const int* dst = EI + NE;

    k_deg_init<<<(NN + 255) / 256, 256, 0, stream>>>(dinv);
    k_deg_edge<<<2048, 256, 0, stream>>>(dst, dinv);
    k_dinv<<<(NN + 255) / 256, 256, 0, stream>>>(dinv);

    k_gemm1<<<(NTILES + 7) / 8, 256, 0, stream>>>(X, W1, H);

    k_zero<<<(NN * HD + 255) / 256, 256, 0, stream>>>(AGG1, NN * HD);
    k_scatter1<<<8192, 256, 0, stream>>>(src, dst, dinv, H, AGG1);
    k_relu_bias<<<(NN * HD + 255) / 256, 256, 0, stream>>>(AGG1, H, dinv, b1);

    k_gemm2<<<(NN + 255) / 256, 256, 0, stream>>>(AGG1, W2, G);
    k_zero<<<(NN * NC + 255) / 256, 256, 0, stream>>>(AGG2, NN * NC);
    k_scatter2<<<(NE + 255) / 256, 256, 0, stream>>>(src, dst, dinv, G, AGG2);
    k_out<<<(NN + 255) / 256, 256, 0, stream>>>(AGG2, G, dinv, b2, out);
}